// OESM_CrossEntropy_41970420417164
// MI455X (gfx1250) — compile-verified
//
#include <hip/hip_runtime.h>
#include <hip/hip_bf16.h>

typedef float v4f __attribute__((ext_vector_type(4)));
typedef int   v4i __attribute__((ext_vector_type(4)));

#define AS1 __attribute__((address_space(1)))
#define AS3 __attribute__((address_space(3)))

// gfx1250 async global->LDS builtins (ASYNCcnt path), probe-confirmed in
// round 1/2. Fallback: vectorized NT b128 loads + global_prefetch_b8.
#if defined(__has_builtin)
#if __has_builtin(__builtin_amdgcn_global_load_async_to_lds_b128) && \
    __has_builtin(__builtin_amdgcn_s_wait_asynccnt)
#define USE_ASYNC_LDS 1
#endif
#endif
#ifndef USE_ASYNC_LDS
#define USE_ASYNC_LDS 0
#endif

__device__ __forceinline__ void online_merge(float& m, float& s, float cm, float e) {
  if (cm > m) { s = s * __expf(m - cm) + e; m = cm; }
  else        { s += e * __expf(cm - m); }
}

__device__ __forceinline__ void acc_vec4(float& m, float& s, v4f v) {
  float cm = fmaxf(fmaxf(v.x, v.y), fmaxf(v.z, v.w));
  float e  = __expf(v.x - cm) + __expf(v.y - cm) +
             __expf(v.z - cm) + __expf(v.w - cm);
  online_merge(m, s, cm, e);
}

// ---------------------------------------------------------------------------
// Kernel 1: per-row logsumexp - x[target]  (one 256-thread block per row)
// 4-stage, 2x-b128-per-stage async pipeline: 3 stages (3KB/wave) in flight.
// ---------------------------------------------------------------------------
__global__ __launch_bounds__(256)
void ce_row_losses(const float* __restrict__ x, const int* __restrict__ tgt,
                   float* __restrict__ losses, int C) {
  const int row = blockIdx.x;
  const int tid = threadIdx.x;
  const float* rowp = x + (size_t)row * (size_t)C;
  const v4f* rowv = (const v4f*)rowp;
  const int nvec = C >> 2;

  float m = -3.402823466e38f;
  float s = 0.0f;

#if USE_ASYNC_LDS
  // Stage = 512 vec4 (2 per lane). Each lane owns its two 16B LDS slots, so
  // only s_wait_asynccnt (per-wave, in-order) is needed — no barriers.
  const int nStage = (nvec + 511) >> 9;
  __shared__ v4f tile[4][2][256];

#define ISSUE_STAGE(S)                                                        \
  do {                                                                        \
    int _s = (S);                                                             \
    int _i0 = (_s << 9) + tid;                                                \
    int _i1 = _i0 + 256;                                                      \
    if (_i0 < nvec)                                                           \
      __builtin_amdgcn_global_load_async_to_lds_b128(                         \
          (AS1 v4i*)(rowv + _i0), (AS3 v4i*)&tile[_s & 3][0][tid], 0, 0);     \
    if (_i1 < nvec)                                                           \
      __builtin_amdgcn_global_load_async_to_lds_b128(                         \
          (AS1 v4i*)(rowv + _i1), (AS3 v4i*)&tile[_s & 3][1][tid], 0, 0);     \
  } while (0)

  // Prologue: fill 3 stages ahead.
  if (0 < nStage) ISSUE_STAGE(0);
  if (1 < nStage) ISSUE_STAGE(1);
  if (2 < nStage) ISSUE_STAGE(2);

  for (int it = 0; it < nStage; ++it) {
    if (it + 3 < nStage) {
      ISSUE_STAGE(it + 3);
      __builtin_amdgcn_s_wait_asynccnt(6);   // stage `it` (2 oldest) complete
    } else if (it + 2 < nStage) {
      __builtin_amdgcn_s_wait_asynccnt(4);
    } else if (it + 1 < nStage) {
      __builtin_amdgcn_s_wait_asynccnt(2);
    } else {
      __builtin_amdgcn_s_wait_asynccnt(0);
    }
    int i0 = (it << 9) + tid;
    int i1 = i0 + 256;
    if (i0 < nvec) acc_vec4(m, s, tile[it & 3][0][tid]);
    if (i1 < nvec) acc_vec4(m, s, tile[it & 3][1][tid]);
  }
#undef ISSUE_STAGE
#else
  const int nIter = (nvec + 255) >> 8;
  for (int it = 0; it < nIter; ++it) {
    int idx = (it << 8) + tid;
    int pf  = idx + (4 << 8);             // prefetch 4 iterations (16KB) ahead
    if (((it & 3) == 0) && pf < nvec)
      __builtin_prefetch((const void*)(rowv + pf), 0, 0);
    if (idx < nvec) {
      v4f v = __builtin_nontemporal_load(rowv + idx);   // streamed once: NT
      acc_vec4(m, s, v);
    }
  }
#endif

  // scalar tail (C % 4) — not hit for C=32000, kept for generality
  int rem = C & 3;
  if (tid < rem) {
    float v = rowp[((C >> 2) << 2) + tid];
    online_merge(m, s, v, 1.0f);
  }

  __shared__ float rm[256];
  __shared__ float rs[256];
  rm[tid] = m; rs[tid] = s;
  __syncthreads();
  for (int off = 128; off > 0; off >>= 1) {
    if (tid < off) {
      float m2 = rm[tid + off], s2 = rs[tid + off];
      float mm = fmaxf(rm[tid], m2);
      rs[tid]  = rs[tid] * __expf(rm[tid] - mm) + s2 * __expf(m2 - mm);
      rm[tid]  = mm;
    }
    __syncthreads();
  }
  if (tid == 0) {
    float lse = rm[0] + logf(rs[0]);
    losses[row] = lse - rowp[tgt[row]];
  }
}

// ---------------------------------------------------------------------------
// Kernel 2: exact mean of top-k of n losses (single 1024-thread block).
// Radix-select the k-th largest via monotone uint keys in LDS, then
// sum = sum(loss > T) + (k - cnt_gt) * T  (exact tie handling).
// ---------------------------------------------------------------------------
__global__ __launch_bounds__(1024)
void topk_mean(const float* __restrict__ losses, float* __restrict__ out,
               int n, int k) {
  __shared__ unsigned skey[8192];
  __shared__ unsigned redu[1024];
  __shared__ float    redf[1024];
  const int tid = threadIdx.x;

  for (int i = tid; i < n; i += 1024) {
    unsigned b = __float_as_uint(losses[i]);
    skey[i] = b ^ ((b >> 31) ? 0xFFFFFFFFu : 0x80000000u);  // order-preserving
  }
  __syncthreads();

  unsigned prefix = 0, maskKnown = 0;
  int kneed = k;
  for (int bit = 31; bit >= 0; --bit) {
    unsigned b = 1u << bit;
    unsigned cnt = 0;
    for (int i = tid; i < n; i += 1024) {
      unsigned kk = skey[i];
      cnt += (((kk & maskKnown) == prefix) && (kk & b)) ? 1u : 0u;
    }
    redu[tid] = cnt;
    __syncthreads();
    for (int off = 512; off > 0; off >>= 1) {
      if (tid < off) redu[tid] += redu[tid + off];
      __syncthreads();
    }
    unsigned total = redu[0];
    __syncthreads();
    if (total >= (unsigned)kneed) prefix |= b;
    else                          kneed -= (int)total;
    maskKnown |= b;
  }
  const unsigned T = prefix;   // key of the k-th largest loss

  float    ssum = 0.0f;
  unsigned cgt  = 0;
  for (int i = tid; i < n; i += 1024) {
    unsigned kk = skey[i];
    if (kk > T) {
      unsigned b = kk ^ ((kk & 0x80000000u) ? 0x80000000u : 0xFFFFFFFFu);
      ssum += __uint_as_float(b);
      cgt++;
    }
  }
  redf[tid] = ssum; redu[tid] = cgt;
  __syncthreads();
  for (int off = 512; off > 0; off >>= 1) {
    if (tid < off) { redf[tid] += redf[tid + off]; redu[tid] += redu[tid + off]; }
    __syncthreads();
  }
  if (tid == 0) {
    unsigned tb = T ^ ((T & 0x80000000u) ? 0x80000000u : 0xFFFFFFFFu);
    float tf = __uint_as_float(tb);
    float total = redf[0] + (float)(k - (int)redu[0]) * tf;
    out[0] = total / (float)k;
  }
}

// ---------------------------------------------------------------------------
extern "C" void kernel_launch(void* const* d_in, const int* in_sizes, int n_in,
                              void* d_out, int out_size, void* d_ws, size_t ws_size,
                              hipStream_t stream) {
  const float* x   = (const float*)d_in[0];
  const int*   tgt = (const int*)d_in[1];
  const int N = in_sizes[1];             // 8192
  const int C = in_sizes[0] / N;         // 32000
  float* losses = (float*)d_ws;          // N floats of scratch

  const int k1 = (int)(1.0 * N);         // DOWN_K
  const int k2 = (int)(0.6 * k1);        // TOP_K -> 4915

  ce_row_losses<<<N, 256, 0, stream>>>(x, tgt, losses, C);
  topk_mean<<<1, 1024, 0, stream>>>(losses, (float*)d_out, N, k2);
}